// LightGCNSI_61632780698010
// MI455X (gfx1250) — compile-verified
//
#include <hip/hip_runtime.h>
#include <stdint.h>

// Problem constants (match reference)
#define N_USERS_C 200000
#define N_ITEMS_C 100000
#define N_NODES_C 300000
#define NNZ_C     4000000
#define D_C       64

typedef uint32_t u32;
typedef uint32_t u32x4 __attribute__((ext_vector_type(4)));
typedef uint32_t u32x8 __attribute__((ext_vector_type(8)));

// =====================================================================
// Kernel 1: e0 = id_emb + side-info embeddings ; acc = e0 ; x = e0
// Pure streaming, float4 (b128) loads/stores.
// =====================================================================
__global__ void lgcn_init(const int* __restrict__ ug, const int* __restrict__ ua,
                          const int* __restrict__ ic,
                          const float* __restrict__ uw, const float* __restrict__ iw,
                          const float* __restrict__ gw, const float* __restrict__ aw,
                          const float* __restrict__ cw,
                          float* __restrict__ x, float* __restrict__ acc)
{
    u32 tid = blockIdx.x * blockDim.x + threadIdx.x;
    if (tid >= (u32)N_NODES_C * 16u) return;
    u32 n = tid >> 4;
    u32 q = (tid & 15u) * 4u;
    float4 r;
    if (n < (u32)N_USERS_C) {
        const float4 a = *(const float4*)(uw + (size_t)n * D_C + q);
        const float4 b = *(const float4*)(gw + (size_t)ug[n] * D_C + q);
        const float4 c = *(const float4*)(aw + (size_t)ua[n] * D_C + q);
        r = make_float4(a.x + b.x + c.x, a.y + b.y + c.y,
                        a.z + b.z + c.z, a.w + b.w + c.w);
    } else {
        u32 m = n - (u32)N_USERS_C;
        const float4 a = *(const float4*)(iw + (size_t)m * D_C + q);
        const float4 b = *(const float4*)(cw + (size_t)ic[m] * D_C + q);
        r = make_float4(a.x + b.x, a.y + b.y, a.z + b.z, a.w + b.w);
    }
    *(float4*)(x   + (size_t)n * D_C + q) = r;
    *(float4*)(acc + (size_t)n * D_C + q) = r;
}

// =====================================================================
// Kernel 2a: COO SpMM using the Tensor Data Mover in GATHER mode.
// Each wave processes 8 edges per iteration: one tensor_load_to_lds with
// gather_mode=1 / 32-bit indices pulls the 8 random x[col] rows (8 x 256B)
// into this wave's LDS double buffer; TENSORcnt waits pipeline iteration
// i's compute against iteration i+1's DMA.  Scatter = f32 atomics (L2 RMW).
// =====================================================================
#define TDM_EDGES 8
#define TDM_BUF_FLOATS (TDM_EDGES * D_C)          // 512 floats = 2048 B
#define SPMM_BLOCK 256
#define SPMM_WAVES (SPMM_BLOCK / 32)

__global__ void lgcn_spmm_tdm(const float* __restrict__ x,
                              float* __restrict__ y,
                              const int* __restrict__ rows,
                              const int* __restrict__ cols,
                              const float* __restrict__ vals)
{
    __shared__ float lds[SPMM_WAVES * 2 * TDM_BUF_FLOATS];   // 32 KB

    const u32 lane = threadIdx.x & 31u;
    const u32 wib  = threadIdx.x >> 5;                       // wave in block
    const u32 wpb  = blockDim.x >> 5;
    const u32 gwid = (u32)__builtin_amdgcn_readfirstlane((int)(blockIdx.x * wpb + wib));
    const u32 GW   = gridDim.x * wpb;
    const u32 nb   = NNZ_C / TDM_EDGES;                      // 500000 full batches

    float* buf0 = &lds[wib * 2u * TDM_BUF_FLOATS];
    float* buf1 = buf0 + TDM_BUF_FLOATS;
    const u32 la0 = (u32)(uintptr_t)buf0;  // low 32 bits of shared addr = LDS offset
    const u32 la1 = (u32)(uintptr_t)buf1;
    const uint64_t ga = (uint64_t)(uintptr_t)x;

    // Issue one TDM gather of 8 rows of x into LDS at 'la'.
    auto issue = [&](u32 b, u32 la) {
        const u32 eb = b * TDM_EDGES;
        // wave-uniform column indices -> SGPRs for the descriptor
        u32 i0 = (u32)__builtin_amdgcn_readfirstlane(cols[eb + 0]);
        u32 i1 = (u32)__builtin_amdgcn_readfirstlane(cols[eb + 1]);
        u32 i2 = (u32)__builtin_amdgcn_readfirstlane(cols[eb + 2]);
        u32 i3 = (u32)__builtin_amdgcn_readfirstlane(cols[eb + 3]);
        u32 i4 = (u32)__builtin_amdgcn_readfirstlane(cols[eb + 4]);
        u32 i5 = (u32)__builtin_amdgcn_readfirstlane(cols[eb + 5]);
        u32 i6 = (u32)__builtin_amdgcn_readfirstlane(cols[eb + 6]);
        u32 i7 = (u32)__builtin_amdgcn_readfirstlane(cols[eb + 7]);
        // D# group 0: count=1, gather_mode=1, 32-bit indices, lds_addr,
        //             global_addr (57b), type=2
        u32x4 g0 = { 0xC0000001u,
                     la,
                     (u32)ga,
                     (u32)((ga >> 32) & 0x01FFFFFFu) | (2u << 30) };
        // D# group 1: data_size=4B, tensor_dim0=64, tensor_dim1=N (OOB bound),
        //             tile_dim0=64, tile_dim1=#indices=8, dim0_stride=64
        u32x8 g1 = { 0x00020000u,                    // data_size=2 (4B), mask=0
                     ((u32)D_C) << 16,               // tensor_dim0 @ bits[79:48] (lo)
                     ((u32)N_NODES_C & 0xFFFFu) << 16, // tensor_dim1 @ [111:80] (lo)
                     ((u32)N_NODES_C >> 16) | (((u32)D_C) << 16), // dim1 hi | tile_dim0
                     (u32)TDM_EDGES,                 // tile_dim1 = #valid indices
                     (u32)D_C,                       // tensor_dim0_stride
                     0u, 0u };
        // D# groups 2/3: 8 x 32-bit row indices
        u32x4 g2 = { i0, i1, i2, i3 };
        u32x4 g3 = { i4, i5, i6, i7 };
        asm volatile("tensor_load_to_lds %0, %1, %2, %3"
                     :: "s"(g0), "s"(g1), "s"(g2), "s"(g3)
                     : "memory");
    };

    u32 b = gwid;
    if (b < nb) issue(b, la0);
    u32 cur = 0;
    for (; b < nb; b += GW, cur ^= 1u) {
        const u32 nxt = b + GW;
        if (nxt < nb) issue(nxt, cur ? la0 : la1);

        // this lane: edge = lane>>2 within batch, quarter-row = lane&3
        const u32 e  = b * TDM_EDGES + (lane >> 2);
        const int   r = rows[e];
        const float v = vals[e];

        if (nxt < nb) { asm volatile("s_wait_tensorcnt 0x1" ::: "memory"); }
        else          { asm volatile("s_wait_tensorcnt 0x0" ::: "memory"); }

        const u32 slot = wib * 2u * TDM_BUF_FLOATS + (cur ? TDM_BUF_FLOATS : 0u)
                       + (lane >> 2) * D_C + (lane & 3u) * 16u;
        float* dst = y + (size_t)r * D_C + (lane & 3u) * 16u;
        #pragma unroll
        for (int k = 0; k < 4; ++k) {
            const float4 xv = *(const float4*)&lds[slot + k * 4];
            atomicAdd(dst + k * 4 + 0, v * xv.x);
            atomicAdd(dst + k * 4 + 1, v * xv.y);
            atomicAdd(dst + k * 4 + 2, v * xv.z);
            atomicAdd(dst + k * 4 + 3, v * xv.w);
        }
    }
}

// =====================================================================
// Kernel 2b: same SpMM via per-lane async global->LDS B128 loads
// (ASYNCcnt path).  2 edges per wave per iteration: 16 lanes x 16B per edge.
// =====================================================================
#define A_EDGES 2
#define A_BUF_FLOATS (A_EDGES * D_C)              // 128 floats = 512 B

__global__ void lgcn_spmm_async(const float* __restrict__ x,
                                float* __restrict__ y,
                                const int* __restrict__ rows,
                                const int* __restrict__ cols,
                                const float* __restrict__ vals)
{
    __shared__ float lds[SPMM_WAVES * 2 * A_BUF_FLOATS];     // 8 KB

    const u32 lane = threadIdx.x & 31u;
    const u32 wib  = threadIdx.x >> 5;
    const u32 wpb  = blockDim.x >> 5;
    const u32 gwid = (u32)__builtin_amdgcn_readfirstlane((int)(blockIdx.x * wpb + wib));
    const u32 GW   = gridDim.x * wpb;
    const u32 nb   = NNZ_C / A_EDGES;                        // 2,000,000 full batches

    const u32 sub = lane >> 4;    // which edge of the pair
    const u32 ch  = lane & 15u;   // which 16B chunk of the 256B row

    float* buf0 = &lds[wib * 2u * A_BUF_FLOATS];
    const u32 la0 = (u32)(uintptr_t)buf0;

    auto issue = [&](u32 b, u32 whichBuf) {
        const u32 e    = b * A_EDGES + sub;
        const u32 c    = (u32)cols[e];
        const u32 voff = c * (u32)(D_C * 4) + ch * 16u;                 // byte offset into x
        const u32 la   = la0 + whichBuf * (A_BUF_FLOATS * 4u)
                       + sub * (D_C * 4u) + ch * 16u;                   // LDS byte address
        asm volatile("global_load_async_to_lds_b128 %0, %1, %2"
                     :: "v"(la), "v"(voff), "s"(x)
                     : "memory");
    };

    u32 b = gwid;
    if (b < nb) issue(b, 0u);
    u32 cur = 0;
    for (; b < nb; b += GW, cur ^= 1u) {
        const u32 nxt = b + GW;
        if (nxt < nb) issue(nxt, cur ^ 1u);

        const u32 e  = b * A_EDGES + sub;
        const int   r = rows[e];
        const float v = vals[e];

        if (nxt < nb) { asm volatile("s_wait_asynccnt 0x1" ::: "memory"); }
        else          { asm volatile("s_wait_asynccnt 0x0" ::: "memory"); }

        const u32 slot = wib * 2u * A_BUF_FLOATS + (cur ? A_BUF_FLOATS : 0u)
                       + sub * D_C + ch * 4u;
        const float4 xv = *(const float4*)&lds[slot];
        float* dst = y + (size_t)r * D_C + ch * 4u;
        atomicAdd(dst + 0, v * xv.x);
        atomicAdd(dst + 1, v * xv.y);
        atomicAdd(dst + 2, v * xv.z);
        atomicAdd(dst + 3, v * xv.w);
    }
}

// =====================================================================
// Kernel 3: acc = (acc + y) * post   (post = 1 for layers 1-2, 0.25 last)
// =====================================================================
__global__ void lgcn_acc(const float* __restrict__ yv, float* __restrict__ acc, float post)
{
    u32 tid = blockIdx.x * blockDim.x + threadIdx.x;
    if (tid >= (u32)N_NODES_C * 16u) return;
    const size_t o = (size_t)tid * 4u;
    const float4 a = *(const float4*)(acc + o);
    const float4 b = *(const float4*)(yv + o);
    *(float4*)(acc + o) = make_float4((a.x + b.x) * post, (a.y + b.y) * post,
                                      (a.z + b.z) * post, (a.w + b.w) * post);
}

// =====================================================================
extern "C" void kernel_launch(void* const* d_in, const int* in_sizes, int n_in,
                              void* d_out, int out_size, void* d_ws, size_t ws_size,
                              hipStream_t stream)
{
    (void)in_sizes; (void)n_in; (void)out_size; (void)ws_size;
    const int*   rows = (const int*)  d_in[0];
    const int*   cols = (const int*)  d_in[1];
    const float* vals = (const float*)d_in[2];
    const int*   ug   = (const int*)  d_in[3];
    const int*   ua   = (const int*)  d_in[4];
    const int*   ic   = (const int*)  d_in[5];
    const float* uw   = (const float*)d_in[6];
    const float* iw   = (const float*)d_in[7];
    const float* gw   = (const float*)d_in[8];
    const float* aw   = (const float*)d_in[9];
    const float* cw   = (const float*)d_in[10];

    float* acc = (float*)d_out;                        // [N_NODES, 64] = outputs concat
    const size_t bufElems = (size_t)N_NODES_C * D_C;   // 19.2M floats = 76.8 MB
    float* bufA = (float*)d_ws;
    float* bufB = bufA + bufElems;

    const dim3 blk(SPMM_BLOCK);
    const u32  nE16 = (u32)N_NODES_C * 16u;
    const dim3 eGrid((nE16 + SPMM_BLOCK - 1) / SPMM_BLOCK);
    const int  SPMM_GRID = 2048;                       // 16384 waves, grid-stride

    // x0 / acc
    lgcn_init<<<eGrid, blk, 0, stream>>>(ug, ua, ic, uw, iw, gw, aw, cw, bufA, acc);

    // layer 1 (TDM gather path): bufB = A * bufA ; acc += bufB
    hipMemsetAsync(bufB, 0, bufElems * sizeof(float), stream);
    lgcn_spmm_tdm<<<SPMM_GRID, blk, 0, stream>>>(bufA, bufB, rows, cols, vals);
    lgcn_acc<<<eGrid, blk, 0, stream>>>(bufB, acc, 1.0f);

    // layer 2 (async global->LDS path): bufA = A * bufB ; acc += bufA
    hipMemsetAsync(bufA, 0, bufElems * sizeof(float), stream);
    lgcn_spmm_async<<<SPMM_GRID, blk, 0, stream>>>(bufB, bufA, rows, cols, vals);
    lgcn_acc<<<eGrid, blk, 0, stream>>>(bufA, acc, 1.0f);

    // layer 3 (TDM gather path): bufB = A * bufA ; acc = (acc + bufB) / 4
    hipMemsetAsync(bufB, 0, bufElems * sizeof(float), stream);
    lgcn_spmm_tdm<<<SPMM_GRID, blk, 0, stream>>>(bufA, bufB, rows, cols, vals);
    lgcn_acc<<<eGrid, blk, 0, stream>>>(bufB, acc, 0.25f);
}